// P_44066364457532
// MI455X (gfx1250) — compile-verified
//
#include <hip/hip_runtime.h>

// ---------------------------------------------------------------------------
// Piola stress P = dW/dF for W(F) = 8*I1 + 10*J^2 - 56*log J + 0.2*(I4^2+I5^2) - 44
// Closed-form gradient; bandwidth-bound elementwise kernel over N 3x3 matrices.
// gfx1250 path: async global<->LDS staging (ASYNCcnt) for coalesced b128 traffic.
// ---------------------------------------------------------------------------

#define THREADS            256
#define TILE_SAMPLES       1024                    // samples per block tile
#define SAMPLES_PER_THREAD (TILE_SAMPLES / THREADS)  // 4
#define TILE_FLOATS        (TILE_SAMPLES * 9)      // 9216 floats = 36 KB LDS
#define TILE_VEC4          (TILE_FLOATS / 4)       // 2304
#define VEC4_PER_THREAD    (TILE_VEC4 / THREADS)   // 9

#if __has_builtin(__builtin_amdgcn_global_load_async_to_lds_b128) && \
    __has_builtin(__builtin_amdgcn_global_store_async_from_lds_b128)
#define USE_ASYNC_LDS 1
#else
#define USE_ASYNC_LDS 0
#endif

// The builtins take pointers to 16-byte int vectors: global (AS1) first, LDS (AS3) second.
typedef int v4i __attribute__((ext_vector_type(4)));
typedef __attribute__((address_space(1))) v4i* gv4i_p;
typedef __attribute__((address_space(3))) v4i* lv4i_p;

__device__ __forceinline__ void wait_async0() {
#if __has_builtin(__builtin_amdgcn_s_wait_asynccnt)
    __builtin_amdgcn_s_wait_asynccnt(0);
#else
    asm volatile("s_wait_asynccnt 0" ::: "memory");
#endif
}

// Analytic gradient of the strain energy for one sample.
// f, p are row-major 3x3: f[3*r + c].
__device__ __forceinline__ void grad_sample(const float f[9], float p[9]) {
    const float g0 = 4.0f, g1 = 0.5f, g2 = 0.5f;

    const float F00 = f[0], F01 = f[1], F02 = f[2];
    const float F10 = f[3], F11 = f[4], F12 = f[5];
    const float F20 = f[6], F21 = f[7], F22 = f[8];

    // C = F^T F (symmetric)
    const float C00 = F00*F00 + F10*F10 + F20*F20;
    const float C11 = F01*F01 + F11*F11 + F21*F21;
    const float C22 = F02*F02 + F12*F12 + F22*F22;
    const float C01 = F00*F01 + F10*F11 + F20*F21;
    const float C02 = F00*F02 + F10*F12 + F20*F22;
    const float C12 = F01*F02 + F11*F12 + F21*F22;

    // cofactor matrix of F (dJ/dF)
    const float cf00 =  (F11*F22 - F12*F21);
    const float cf01 = -(F10*F22 - F12*F20);
    const float cf02 =  (F10*F21 - F11*F20);
    const float cf10 = -(F01*F22 - F02*F21);
    const float cf11 =  (F00*F22 - F02*F20);
    const float cf12 = -(F00*F21 - F01*F20);
    const float cf20 =  (F01*F12 - F02*F11);
    const float cf21 = -(F00*F12 - F02*F10);
    const float cf22 =  (F00*F11 - F01*F10);

    const float J = F00*cf00 + F01*cf01 + F02*cf02;

    // B = adj(C) = J^2 * C^{-1} (symmetric)
    const float B00 = C11*C22 - C12*C12;
    const float B11 = C00*C22 - C02*C02;
    const float B22 = C00*C11 - C01*C01;
    const float B01 = C02*C12 - C01*C22;
    const float B02 = C01*C12 - C02*C11;
    const float B12 = C01*C02 - C00*C12;

    const float I4 = g0*C00 + g1*C11 + g2*C22;
    const float I5 = g0*B00 + g1*B11 + g2*B22;

    const float invJ     = 1.0f / J;           // single division per sample
    const float inv_detC = invJ * invJ;        // det(C) = J^2
    const float cscale   = 20.0f*J - 56.0f*invJ;

    // BGB (symmetric): BGB_ij = g0*Bi0*Bj0 + g1*Bi1*Bj1 + g2*Bi2*Bj2
    const float BGB00 = g0*B00*B00 + g1*B01*B01 + g2*B02*B02;
    const float BGB01 = g0*B00*B01 + g1*B01*B11 + g2*B02*B12;
    const float BGB02 = g0*B00*B02 + g1*B01*B12 + g2*B02*B22;
    const float BGB11 = g0*B01*B01 + g1*B11*B11 + g2*B12*B12;
    const float BGB12 = g0*B01*B02 + g1*B11*B12 + g2*B12*B22;
    const float BGB22 = g0*B02*B02 + g1*B12*B12 + g2*B22*B22;

    // K = 0.8*I5*(I5*B - BGB)/detC (symmetric)
    const float s5  = 0.8f * I5 * inv_detC;
    const float K00 = s5 * (I5*B00 - BGB00);
    const float K01 = s5 * (I5*B01 - BGB01);
    const float K02 = s5 * (I5*B02 - BGB02);
    const float K11 = s5 * (I5*B11 - BGB11);
    const float K12 = s5 * (I5*B12 - BGB12);
    const float K22 = s5 * (I5*B22 - BGB22);

    // column scales: 16 (from I1 term) + 0.8*I4*g_j (from I4 term)
    const float s4 = 0.8f * I4;
    const float a0 = 16.0f + s4*g0;
    const float a1 = 16.0f + s4*g1;
    const float a2 = 16.0f + s4*g2;

    // P = F*diag(a) + F*K + cscale*cof(F)
    p[0] = a0*F00 + F00*K00 + F01*K01 + F02*K02 + cscale*cf00;
    p[1] = a1*F01 + F00*K01 + F01*K11 + F02*K12 + cscale*cf01;
    p[2] = a2*F02 + F00*K02 + F01*K12 + F02*K22 + cscale*cf02;
    p[3] = a0*F10 + F10*K00 + F11*K01 + F12*K02 + cscale*cf10;
    p[4] = a1*F11 + F10*K01 + F11*K11 + F12*K12 + cscale*cf11;
    p[5] = a2*F12 + F10*K02 + F11*K12 + F12*K22 + cscale*cf12;
    p[6] = a0*F20 + F20*K00 + F21*K01 + F22*K02 + cscale*cf20;
    p[7] = a1*F21 + F20*K01 + F21*K11 + F22*K12 + cscale*cf21;
    p[8] = a2*F22 + F20*K02 + F21*K12 + F22*K22 + cscale*cf22;
}

__global__ __launch_bounds__(THREADS)
void piola_grad_kernel(const float* __restrict__ Fg, float* __restrict__ Pg, int nsamples) {
    __shared__ float lds[TILE_FLOATS];   // 36 KB: F tile in, P tile out (reused)

    const int       tid       = threadIdx.x;
    const long long tileBase  = (long long)blockIdx.x * TILE_SAMPLES;   // sample index
    const long long floatBase = tileBase * 9;

    if (tileBase + TILE_SAMPLES <= (long long)nsamples) {
        // ---- fast path: full tile, staged through LDS -------------------
#if USE_ASYNC_LDS
        gv4i_p gsrc = (gv4i_p)(Fg + floatBase);   // 16-byte aligned (tile is 36864B)
        lv4i_p lbas = (lv4i_p)lds;
#pragma unroll
        for (int i = 0; i < VEC4_PER_THREAD; ++i) {
            const int idx = tid + i * THREADS;    // float4 chunk index
            __builtin_amdgcn_global_load_async_to_lds_b128(gsrc + idx, lbas + idx, 0, 0);
        }
        wait_async0();            // this wave's async loads landed in LDS
#else
        const float4* gsrc4 = (const float4*)(Fg + floatBase);
        float4*       l4    = (float4*)lds;
#pragma unroll
        for (int i = 0; i < VEC4_PER_THREAD; ++i) {
            const int idx = tid + i * THREADS;
            l4[idx] = gsrc4[idx];
        }
#endif
        __syncthreads();          // all waves' tile data visible

        // strided sample mapping: lane t reads floats at 9*t + ... ->
        // stride 9 vs 64 banks (coprime) => conflict-free scalar LDS access
#pragma unroll
        for (int j = 0; j < SAMPLES_PER_THREAD; ++j) {
            const int s = tid + j * THREADS;  // sample index within tile
            float f[9], pr[9];
#pragma unroll
            for (int k = 0; k < 9; ++k) f[k] = lds[s * 9 + k];
            grad_sample(f, pr);
            // this thread exclusively owns sample s's LDS slots
#pragma unroll
            for (int k = 0; k < 9; ++k) lds[s * 9 + k] = pr[k];
        }
        __syncthreads();          // all P values written to LDS

#if USE_ASYNC_LDS
        gv4i_p gdst = (gv4i_p)(Pg + floatBase);
        lv4i_p lout = (lv4i_p)lds;
#pragma unroll
        for (int i = 0; i < VEC4_PER_THREAD; ++i) {
            const int idx = tid + i * THREADS;
            __builtin_amdgcn_global_store_async_from_lds_b128(gdst + idx, lout + idx, 0, 0);
        }
        // ASYNCcnt drained by the implicit wait-idle at s_endpgm
#else
        float4* gdst4 = (float4*)(Pg + floatBase);
        float4* l4o   = (float4*)lds;
#pragma unroll
        for (int i = 0; i < VEC4_PER_THREAD; ++i) {
            const int idx = tid + i * THREADS;
            gdst4[idx] = l4o[idx];
        }
#endif
    } else {
        // ---- tail tile: guarded per-sample path -------------------------
        for (int j = 0; j < SAMPLES_PER_THREAD; ++j) {
            const long long s = tileBase + tid + j * THREADS;
            if (s < (long long)nsamples) {
                float f[9], pr[9];
                const float* src = Fg + s * 9;
#pragma unroll
                for (int k = 0; k < 9; ++k) f[k] = src[k];
                grad_sample(f, pr);
                float* dst = Pg + s * 9;
#pragma unroll
                for (int k = 0; k < 9; ++k) dst[k] = pr[k];
            }
        }
    }
}

extern "C" void kernel_launch(void* const* d_in, const int* in_sizes, int n_in,
                              void* d_out, int out_size, void* d_ws, size_t ws_size,
                              hipStream_t stream) {
    (void)n_in; (void)out_size; (void)d_ws; (void)ws_size;
    const float* F = (const float*)d_in[0];
    float*       P = (float*)d_out;
    const int nsamples = in_sizes[0] / 9;                 // [N,3,3] flattened
    const int blocks   = (nsamples + TILE_SAMPLES - 1) / TILE_SAMPLES;
    piola_grad_kernel<<<blocks, THREADS, 0, stream>>>(F, P, nsamples);
}